// LieConv_36180804501846
// MI455X (gfx1250) — compile-verified
//
#include <hip/hip_runtime.h>
#include <hip/hip_bf16.h>
#include <stdint.h>

// Problem constants (match reference)
#define BS  4
#define NPT 2048
#define D   4
#define CI  64
#define KNN 32
#define CM  16
#define CO  64
#define HID 32

typedef __attribute__((ext_vector_type(2))) float v2f;
typedef __attribute__((ext_vector_type(8))) float v8f;

// CDNA5 native fp32 WMMA: D[16x16] = A[16x4] * B[4x16] + C
// A layout: lanes 0-15 hold M=0..15 with K={0,1}; lanes 16-31 hold M=0..15 with K={2,3}
// B layout: VGPR0 = row K=koff, VGPR1 = row K=koff+1 (koff = lane<16 ? 0 : 2), col = lane&15
// C/D layout: VGPR r -> M = r (lanes 0-15) / r+8 (lanes 16-31), N = lane&15 (per docs)
__device__ __forceinline__ v8f wmma4(v2f a, v2f b, v8f c) {
  return __builtin_amdgcn_wmma_f32_16x16x4_f32(
      /*neg_a=*/false, a, /*neg_b=*/false, b,
      /*c_mod=*/(short)0, c, /*reuse_a=*/false, /*reuse_b=*/false);
}

// swish via hardware v_exp_f32 + v_rcp_f32 (avoids IEEE div_scale/newton chain
// that showed up in the previous build's assembly; ~1ulp rcp is plenty here)
__device__ __forceinline__ float swish1(float x) {
  return x * __builtin_amdgcn_rcpf(1.0f + __expf(-x));
}
__device__ __forceinline__ v8f swish8(v8f x) {
  v8f y;
#pragma unroll
  for (int r = 0; r < 8; ++r) y[r] = swish1(x[r]);
  return y;
}
__device__ __forceinline__ v8f bcast8(float v) {
  v8f y;
#pragma unroll
  for (int r = 0; r < 8; ++r) y[r] = v;
  return y;
}

// ---------------------------------------------------------------------------
// Kernel 1: distances + iterative top-K (K=32 smallest, stable tie -> low idx)
// one block per query point (b,n); 256 threads
// ---------------------------------------------------------------------------
__global__ void k1_topk(const float* __restrict__ abq,
                        const unsigned char* __restrict__ mask,
                        int* __restrict__ nidx) {
  __shared__ float sdist[NPT];
  __shared__ float rmin[256];
  __shared__ int   ridx[256];

  const int tid = threadIdx.x;
  const int q   = blockIdx.x;          // b*NPT + n
  const int b   = q / NPT;

  // distance row: ||abq[b,n,col,:]|| with mask -> 1e8
  const float4* row = (const float4*)(abq + (size_t)q * NPT * D);
  for (int col = tid; col < NPT; col += 256) {
    float4 v = row[col];
    float d = sqrtf(v.x * v.x + v.y * v.y + v.z * v.z + v.w * v.w);
    if (!mask[b * NPT + col]) d = 1.0e8f;
    sdist[col] = d;
  }
  __syncthreads();

  for (int k = 0; k < KNN; ++k) {
    float best = 3.0e38f;
    int   bi   = 0x7fffffff;
    for (int col = tid; col < NPT; col += 256) {
      float v = sdist[col];
      if (v < best) { best = v; bi = col; }   // ascending col -> stable tie
    }
    rmin[tid] = best;
    ridx[tid] = bi;
    __syncthreads();
    for (int s = 128; s > 0; s >>= 1) {
      if (tid < s) {
        float v2 = rmin[tid + s]; int i2 = ridx[tid + s];
        if (v2 < rmin[tid] || (v2 == rmin[tid] && i2 < ridx[tid])) {
          rmin[tid] = v2; ridx[tid] = i2;
        }
      }
      __syncthreads();
    }
    if (tid == 0) {
      int sel = ridx[0];
      nidx[(size_t)q * KNN + k] = sel;
      sdist[sel] = 3.0e38f;              // remove from further rounds
    }
    __syncthreads();
  }
}

// ---------------------------------------------------------------------------
// Kernel 2: gather + WeightNet MLP (WMMA f32) + masked pc einsum (WMMA f32)
// one wave (32 threads) per query point
// ---------------------------------------------------------------------------
__global__ void k2_weightnet_pc(const float* __restrict__ abq,
                                const float* __restrict__ vals,
                                const unsigned char* __restrict__ mask,
                                const float* __restrict__ w1, const float* __restrict__ b1,
                                const float* __restrict__ w2, const float* __restrict__ b2,
                                const float* __restrict__ w3, const float* __restrict__ b3,
                                const int* __restrict__ nidx,
                                float* __restrict__ pcws) {
  __shared__ float abq_s[KNN * D];        // 512 B
  __shared__ float vals_s[KNN * CI];      // 8 KB  (masked vm, rows = neighbor)
  __shared__ float h_s[KNN * HID];        // 4 KB  (activations, reused)
  __shared__ float wm_s[KNN * CM];        // 2 KB  (masked weightnet out)
  __shared__ int   vmask_s[KNN];

  const int lane = threadIdx.x;
  const int q    = blockIdx.x;            // b*NPT + n
  const int b    = q / NPT;
  const int n    = q - b * NPT;

  const int idx   = nidx[(size_t)q * KNN + lane];
  const int valid = mask[b * NPT + n] && mask[b * NPT + idx];
  vmask_s[lane] = valid;

  // gather abq row [4]
  *(float4*)(abq_s + lane * D) =
      *(const float4*)(abq + (((size_t)q) * NPT + idx) * D);
  // gather vals row [64], pre-masked (vm)
  const float* vsrc = vals + ((size_t)b * NPT + idx) * CI;
#pragma unroll
  for (int c = 0; c < CI / 4; ++c) {
    float4 v = *(const float4*)(vsrc + c * 4);
    if (!valid) { v.x = v.y = v.z = v.w = 0.0f; }
    *(float4*)(vals_s + lane * CI + c * 4) = v;
  }
  __syncthreads();

  const int hl   = lane & 15;
  const int koff = (lane < 16) ? 0 : 2;
  const int mofs = (lane < 16) ? 0 : 8;   // D-tile row offset

  // ---- layer 1: [32,4] @ w1[4,32] + b1, swish ----
  v8f acc1[2][2];
#pragma unroll
  for (int mt = 0; mt < 2; ++mt) {
    v2f a;
    a.x = abq_s[(mt * 16 + hl) * D + koff];
    a.y = abq_s[(mt * 16 + hl) * D + koff + 1];
#pragma unroll
    for (int nt = 0; nt < 2; ++nt) {
      int col = nt * 16 + hl;
      v2f bb; bb.x = w1[koff * HID + col]; bb.y = w1[(koff + 1) * HID + col];
      acc1[mt][nt] = swish8(wmma4(a, bb, bcast8(b1[col])));
    }
  }
#pragma unroll
  for (int mt = 0; mt < 2; ++mt)
#pragma unroll
    for (int nt = 0; nt < 2; ++nt)
#pragma unroll
      for (int r = 0; r < 8; ++r)
        h_s[(mt * 16 + r + mofs) * HID + nt * 16 + hl] = acc1[mt][nt][r];
  __syncthreads();

  // ---- layer 2: [32,32] @ w2[32,32] + b2, swish ----
  v8f acc2[2][2];
#pragma unroll
  for (int mt = 0; mt < 2; ++mt)
#pragma unroll
    for (int nt = 0; nt < 2; ++nt)
      acc2[mt][nt] = bcast8(b2[nt * 16 + hl]);
#pragma unroll
  for (int ks = 0; ks < 8; ++ks) {
    int k0 = ks * 4 + koff;
    v2f a[2];
#pragma unroll
    for (int mt = 0; mt < 2; ++mt) {
      int row = mt * 16 + hl;
      a[mt].x = h_s[row * HID + k0];
      a[mt].y = h_s[row * HID + k0 + 1];
    }
#pragma unroll
    for (int nt = 0; nt < 2; ++nt) {
      int col = nt * 16 + hl;
      v2f bb; bb.x = w2[k0 * HID + col]; bb.y = w2[(k0 + 1) * HID + col];
#pragma unroll
      for (int mt = 0; mt < 2; ++mt)
        acc2[mt][nt] = wmma4(a[mt], bb, acc2[mt][nt]);
    }
  }
  __syncthreads();   // all reads of h_s done before overwrite
#pragma unroll
  for (int mt = 0; mt < 2; ++mt)
#pragma unroll
    for (int nt = 0; nt < 2; ++nt) {
      v8f s = swish8(acc2[mt][nt]);
#pragma unroll
      for (int r = 0; r < 8; ++r)
        h_s[(mt * 16 + r + mofs) * HID + nt * 16 + hl] = s[r];
    }
  __syncthreads();

  // ---- layer 3: [32,32] @ w3[32,16] + b3, swish, neighbor mask ----
  v8f acc3[2];
#pragma unroll
  for (int mt = 0; mt < 2; ++mt) acc3[mt] = bcast8(b3[hl]);
#pragma unroll
  for (int ks = 0; ks < 8; ++ks) {
    int k0 = ks * 4 + koff;
    v2f bb; bb.x = w3[k0 * CM + hl]; bb.y = w3[(k0 + 1) * CM + hl];
#pragma unroll
    for (int mt = 0; mt < 2; ++mt) {
      int row = mt * 16 + hl;
      v2f a; a.x = h_s[row * HID + k0]; a.y = h_s[row * HID + k0 + 1];
      acc3[mt] = wmma4(a, bb, acc3[mt]);
    }
  }
#pragma unroll
  for (int mt = 0; mt < 2; ++mt) {
    v8f s = swish8(acc3[mt]);
#pragma unroll
    for (int r = 0; r < 8; ++r) {
      int m = mt * 16 + r + mofs;
      wm_s[m * CM + hl] = vmask_s[m] ? s[r] : 0.0f;
    }
  }
  __syncthreads();

  // ---- pc: vm^T [64,32] @ wm [32,16] -> [64,16] ----
  v8f pc[4];
#pragma unroll
  for (int mt = 0; mt < 4; ++mt) pc[mt] = bcast8(0.0f);
#pragma unroll
  for (int ks = 0; ks < 8; ++ks) {
    int k0 = ks * 4 + koff;                       // neighbor index
    v2f bb; bb.x = wm_s[k0 * CM + hl]; bb.y = wm_s[(k0 + 1) * CM + hl];
#pragma unroll
    for (int mt = 0; mt < 4; ++mt) {
      int ci = mt * 16 + hl;
      v2f a; a.x = vals_s[k0 * CI + ci]; a.y = vals_s[(k0 + 1) * CI + ci];
      pc[mt] = wmma4(a, bb, pc[mt]);
    }
  }
  // store pc flat: index = ci*CM + cm (matches reshape(bs,n,CI*CM))
  float* dst = pcws + (size_t)q * (CI * CM);
#pragma unroll
  for (int mt = 0; mt < 4; ++mt)
#pragma unroll
    for (int r = 0; r < 8; ++r) {
      int ci = mt * 16 + r + mofs;
      dst[ci * CM + hl] = pc[mt][r];
    }
}

// ---------------------------------------------------------------------------
// Kernel 3: out = pc [16q x 1024] @ Wl [1024 x 64] + bl, masked.
// one wave per 16 queries; pc tile staged in 64 KB LDS
// ---------------------------------------------------------------------------
__global__ void k3_final(const float* __restrict__ pcws,
                         const float* __restrict__ Wl,
                         const float* __restrict__ bl,
                         const unsigned char* __restrict__ mask,
                         float* __restrict__ out) {
  __shared__ float pct[16 * CI * CM];     // 64 KB
  const int lane = threadIdx.x;
  const size_t q0 = (size_t)blockIdx.x * 16;

  const float4* src = (const float4*)(pcws + q0 * (CI * CM));
  float4* dstl = (float4*)pct;
  for (int i = lane; i < 16 * (CI * CM) / 4; i += 32) dstl[i] = src[i];
  __syncthreads();

  const int hl   = lane & 15;
  const int koff = (lane < 16) ? 0 : 2;
  const int mofs = (lane < 16) ? 0 : 8;

  v8f acc[4];
#pragma unroll
  for (int nt = 0; nt < 4; ++nt) acc[nt] = bcast8(bl[nt * 16 + hl]);

  for (int ks = 0; ks < (CI * CM) / 4; ++ks) {     // 256 k-steps
    int k0 = ks * 4 + koff;
    v2f a; a.x = pct[hl * (CI * CM) + k0]; a.y = pct[hl * (CI * CM) + k0 + 1];
#pragma unroll
    for (int nt = 0; nt < 4; ++nt) {
      int col = nt * 16 + hl;
      v2f bb;
      bb.x = Wl[(size_t)k0 * CO + col];
      bb.y = Wl[(size_t)(k0 + 1) * CO + col];
      acc[nt] = wmma4(a, bb, acc[nt]);
    }
  }

#pragma unroll
  for (int nt = 0; nt < 4; ++nt)
#pragma unroll
    for (int r = 0; r < 8; ++r) {
      size_t g = q0 + r + mofs;                    // flat b*NPT+n
      float v = mask[g] ? acc[nt][r] : 0.0f;
      out[g * CO + nt * 16 + hl] = v;
    }
}

// ---------------------------------------------------------------------------
extern "C" void kernel_launch(void* const* d_in, const int* in_sizes, int n_in,
                              void* d_out, int out_size, void* d_ws, size_t ws_size,
                              hipStream_t stream) {
  const float* abq  = (const float*)d_in[0];
  const float* vals = (const float*)d_in[1];
  const unsigned char* mask = (const unsigned char*)d_in[2];  // jnp bool = 1 byte
  const float* w1 = (const float*)d_in[3];
  const float* b1 = (const float*)d_in[4];
  const float* w2 = (const float*)d_in[5];
  const float* b2 = (const float*)d_in[6];
  const float* w3 = (const float*)d_in[7];
  const float* b3 = (const float*)d_in[8];
  const float* Wl = (const float*)d_in[9];
  const float* bl = (const float*)d_in[10];
  float* out = (float*)d_out;

  const int NQ = BS * NPT;                               // 8192 queries
  int*   nidx = (int*)d_ws;                              // 1 MB
  float* pcws = (float*)((char*)d_ws + (size_t)NQ * KNN * sizeof(int)); // 32 MB

  k1_topk<<<NQ, 256, 0, stream>>>(abq, mask, nidx);
  k2_weightnet_pc<<<NQ, 32, 0, stream>>>(abq, vals, mask, w1, b1, w2, b2,
                                         w3, b3, nidx, pcws);
  k3_final<<<NQ / 16, 32, 0, stream>>>(pcws, Wl, bl, mask, out);
}